// GCNLayer_84284438217070
// MI455X (gfx1250) — compile-verified
//
#include <hip/hip_runtime.h>
#include <hip/hip_bf16.h>

typedef __attribute__((ext_vector_type(2))) float v2f;
typedef __attribute__((ext_vector_type(8))) float v8f;

#define DD 128   // input dim
#define OO 128   // output dim
#define RR 3     // relations

// ---------------------------------------------------------------------------
// Zero workspace (agg + deg) with float4 stores.
// ---------------------------------------------------------------------------
__global__ __launch_bounds__(256) void rgcn_zero(float* __restrict__ p, long n) {
    long i  = (long)blockIdx.x * blockDim.x + threadIdx.x;
    long i4 = i * 4;
    if (i4 + 3 < n) {
        ((float4*)p)[i] = float4{0.f, 0.f, 0.f, 0.f};
    } else if (i4 < n) {
        for (long k = i4; k < n; ++k) p[k] = 0.f;
    }
}

// ---------------------------------------------------------------------------
// Scatter: one wave32 per edge. Lane l handles 4 consecutive feature floats
// (float4 coalesced read of x[src], 4 global fp32 atomics into agg[r][dst]).
// Lane 0 bumps the fp32 in-degree counter. All traffic is L2-resident
// (agg 76.8 MB + x 25.6 MB < 192 MB L2).
// ---------------------------------------------------------------------------
__global__ __launch_bounds__(256) void rgcn_scatter(
    const float* __restrict__ x,
    const int*   __restrict__ esrc,
    const int*   __restrict__ edst,
    float* __restrict__ agg,   // [RR, N, DD]
    float* __restrict__ deg,   // [RR, N]
    int N, int E)
{
    const int w    = blockIdx.x * (blockDim.x >> 5) + (threadIdx.x >> 5);
    const int lane = threadIdx.x & 31;
    if (w >= RR * E) return;
    const int r = w / E;
    const int e = w - r * E;

    const int src = esrc[(size_t)r * E + e];
    const int dst = edst[(size_t)r * E + e];

    const float4 v = ((const float4*)(x + (size_t)src * DD))[lane];
    float* o = agg + ((size_t)r * N + dst) * DD + lane * 4;
    atomicAdd(o + 0, v.x);
    atomicAdd(o + 1, v.y);
    atomicAdd(o + 2, v.z);
    atomicAdd(o + 3, v.w);
    if (lane == 0) atomicAdd(deg + (size_t)r * N + dst, 1.0f);
}

// ---------------------------------------------------------------------------
// Normalize: one wave per (relation, node) row: agg /= max(deg, 1).
// ---------------------------------------------------------------------------
__global__ __launch_bounds__(256) void rgcn_norm(
    float* __restrict__ agg, const float* __restrict__ deg, int RN)
{
    const int w    = blockIdx.x * (blockDim.x >> 5) + (threadIdx.x >> 5);
    const int lane = threadIdx.x & 31;
    if (w >= RN) return;
    const float s = 1.0f / fmaxf(deg[w], 1.0f);
    float4* p = (float4*)(agg + (size_t)w * DD);
    float4 v = p[lane];
    v.x *= s; v.y *= s; v.z *= s; v.w *= s;
    p[lane] = v;
}

// ---------------------------------------------------------------------------
// GEMM: h[n,o] = sum_{s=0..3} A_s[n,:] @ W_s[:,o]  (A_s = agg[r] for s<3, x
// for s==3), then + bias, ReLU, * dropout_mask. fp32 WMMA 16x16x4.
//
// Block = 256 threads = 8 waves; each wave owns a 16-row strip x 128 cols
// (8 accumulator tiles of 16x16). Weight segment (128x128 f32 = 64 KB)
// staged in LDS per K-segment in K-PAIR-PACKED layout:
//     wl[(k>>1)*256 + col*2 + (k&1)] = W[k][col]
// so each lane's B fragment (W[kb][col], W[kb+1][col]) is one aligned
// ds_load_b64 directly into the WMMA source register pair (no shuffles).
//
// A layout (16x4 f32, ISA 7.12.2): lane<16 -> row M=lane, K=kk,kk+1 ;
//                                  lane>=16 -> row M=lane-16, K=kk+2,kk+3.
// B layout (4x16 f32, symmetric):  v.x = W[kb][col], v.y = W[kb+1][col],
//                                  kb = kk + 2*(lane>=16), col = tile*16 + (lane&15).
// C/D layout: VGPR i -> row = i + 8*(lane>=16), col = lane&15 (per tile).
// ---------------------------------------------------------------------------
__global__ __launch_bounds__(256) void rgcn_gemm(
    const float* __restrict__ agg,     // [RR, N, DD] normalized
    const float* __restrict__ x,       // [N, DD]
    const float* __restrict__ weight,  // [RR, DD, OO]
    const float* __restrict__ loop_w,  // [DD, OO]
    const float* __restrict__ bias,    // [OO]
    const float* __restrict__ mask,    // [N, OO]
    float* __restrict__ out,           // [N, OO]
    int N)
{
    __shared__ float wl[DD * OO];      // 64 KB: one weight segment, pair-packed

    const int tid  = threadIdx.x;
    const int wave = tid >> 5;
    const int lane = tid & 31;
    const int ln15 = lane & 15;
    const int hi   = lane >> 4;        // 0 or 1
    const int row0 = blockIdx.x * 128 + wave * 16;

    v8f acc[8];
    const v8f vzero = {0.f, 0.f, 0.f, 0.f, 0.f, 0.f, 0.f, 0.f};
#pragma unroll
    for (int j = 0; j < 8; ++j) acc[j] = vzero;

    // clamped A row (per-lane row of the strip); guards stay on loads/stores
    // only so EXEC is all-1s at every WMMA.
    int arow = row0 + ln15;
    if (arow > N - 1) arow = N - 1;

    for (int s = 0; s < 4; ++s) {
        const float* Wseg = (s < 3) ? (weight + (size_t)s * DD * OO) : loop_w;
        const float* A    = (s < 3) ? (agg + (size_t)s * N * DD) : x;

        __syncthreads();  // previous segment's LDS reads done
        // Stage 64 rows-of-pairs: pair p = i>>7 (k = 2p, 2p+1), col = i&127.
        // One ds_store_b64 per element pair; global reads intra-row coalesced.
#pragma unroll
        for (int i = tid; i < (DD / 2) * OO; i += 256) {
            const int kp  = i >> 7;        // k pair index
            const int col = i & (OO - 1);
            v2f pr;
            pr.x = Wseg[(2 * kp)     * OO + col];
            pr.y = Wseg[(2 * kp + 1) * OO + col];
            ((v2f*)wl)[i] = pr;            // wl[kp*256 + col*2 + {0,1}]
        }
        __syncthreads();

        const float* Arow = A + (size_t)arow * DD;
        const v2f* wlp_base = (const v2f*)wl;
#pragma unroll 2
        for (int kk = 0; kk < DD; kk += 4) {
            const int kb = kk + 2 * hi;                 // even
            const v2f a = *(const v2f*)(Arow + kb);     // global_load_b64
            const v2f* wlp = wlp_base + ((kb >> 1) * OO);
#pragma unroll
            for (int j = 0; j < 8; ++j) {
                const int col = j * 16 + ln15;
                const v2f b = wlp[col];                 // ds_load_b64
                acc[j] = __builtin_amdgcn_wmma_f32_16x16x4_f32(
                    false, a, false, b, (short)0, acc[j], false, false);
            }
        }
    }

    // Epilogue: bias + ReLU + dropout mask, guarded stores.
#pragma unroll
    for (int j = 0; j < 8; ++j) {
        const int col  = j * 16 + ln15;
        const float bc = bias[col];
#pragma unroll
        for (int i = 0; i < 8; ++i) {
            const int row = row0 + i + 8 * hi;
            if (row < N) {
                float h = acc[j][i] + bc;
                h = h > 0.f ? h : 0.f;
                out[(size_t)row * OO + col] = h * mask[(size_t)row * OO + col];
            }
        }
    }
}

// ---------------------------------------------------------------------------
extern "C" void kernel_launch(void* const* d_in, const int* in_sizes, int n_in,
                              void* d_out, int out_size, void* d_ws, size_t ws_size,
                              hipStream_t stream) {
    const float* x      = (const float*)d_in[0];
    const float* weight = (const float*)d_in[1];
    const float* loop_w = (const float*)d_in[2];
    const float* bias   = (const float*)d_in[3];
    const float* mask   = (const float*)d_in[4];
    const int*   esrc   = (const int*)d_in[5];
    const int*   edst   = (const int*)d_in[6];

    const int N = in_sizes[0] / DD;       // 50000
    const int E = in_sizes[5] / RR;       // 640000

    float* agg = (float*)d_ws;                       // RR*N*DD floats (76.8 MB)
    float* deg = agg + (size_t)RR * N * DD;          // RR*N floats

    // 1) zero agg + deg
    const long zn = (long)RR * N * DD + (long)RR * N;
    {
        const long t = (zn + 3) / 4;
        rgcn_zero<<<(int)((t + 255) / 256), 256, 0, stream>>>((float*)d_ws, zn);
    }
    // 2) scatter-add (one wave per edge)
    {
        const long waves  = (long)RR * E;
        const int  blocks = (int)((waves + 7) / 8);
        rgcn_scatter<<<blocks, 256, 0, stream>>>(x, esrc, edst, agg, deg, N, E);
    }
    // 3) normalize by in-degree
    {
        const int RN = RR * N;
        rgcn_norm<<<(RN + 7) / 8, 256, 0, stream>>>(agg, deg, RN);
    }
    // 4) fused WMMA GEMM + bias + ReLU + dropout
    {
        const int blocks = (N + 127) / 128;
        rgcn_gemm<<<blocks, 256, 0, stream>>>(agg, x, weight, loop_w, bias, mask,
                                              (float*)d_out, N);
    }
}